// AttentionLayer_48430051229958
// MI455X (gfx1250) — compile-verified
//
#include <hip/hip_runtime.h>

typedef float v2f __attribute__((ext_vector_type(2)));
typedef float v8f __attribute__((ext_vector_type(8)));

#define KNB  32
#define DDIM 128
#define LDIM 64
#define NEG_SLOPE 0.2f

__device__ __forceinline__ float lrelu(float x) {
    // slope < 1  =>  leaky_relu(x) == max(x, slope*x)
    return fmaxf(x, x * NEG_SLOPE);
}

__global__ __launch_bounds__(256) void gat_fused_kernel(
    const float* __restrict__ X,       // [N, 32, 128] para_neighbors
    const float* __restrict__ labels,  // [N, 32, 64]  para_nei_labels
    const float* __restrict__ W,       // [128, 128]   linear (row = reduce dim)
    const float* __restrict__ evec,    // [128]        e_vec
    const float* __restrict__ bias,    // [32]         bias
    float* __restrict__ out,           // [N, 64]
    int N)
{
    // W pre-swizzled into WMMA B-fragment-major order (64 KB):
    // fragment f = j*32+kk occupies 256 B; lane l holds its two B values
    // (rows kk*4 + (l>=16?2:0) + {0,1}, col j*16 + (l&15)) contiguously.
    // -> inner loop B fetch = one ds_load_b64, single base VGPR, imm offset.
    __shared__ float sWf[DDIM * DDIM];

    for (int i = threadIdx.x; i < 8 * 32 * 32; i += blockDim.x) {
        const int l   = i & 31;
        const int kk  = (i >> 5) & 31;
        const int j   = i >> 10;
        const int row = kk * 4 + ((l >> 4) << 1);
        const int col = j * 16 + (l & 15);
        sWf[i * 2 + 0] = W[row * DDIM + col];
        sWf[i * 2 + 1] = W[(row + 1) * DDIM + col];
    }
    __syncthreads();

    const int lane   = threadIdx.x & 31;
    const int laneHi = lane >> 4;              // 0: lanes 0-15, 1: lanes 16-31
    const int m16    = lane & 15;

    const float bias_l = bias[lane];

    const int wid    = (int)((blockIdx.x * blockDim.x + threadIdx.x) >> 5);
    const int nwaves = (int)((gridDim.x * blockDim.x) >> 5);

    for (int n = wid; n < N; n += nwaves) {
        const float* lab = labels + (size_t)n * KNB * LDIM;
        // overlap the label-row HBM fetch with the GEMM below
        __builtin_prefetch(lab + lane * 16, 0, 0);

        float edot = 0.0f;                 // lane k ends with e-dot of neighbor k

        #pragma unroll 1
        for (int half = 0; half < 2; ++half) {
            // ---- A fragments: 16 rows x 128 cols of X, f32 16x16x4 layout ----
            // lane l supplies row m16, K = 4*kk + (laneHi?2:0) + {0,1} -> float2
            const float* xrow =
                X + ((size_t)n * KNB + half * 16 + m16) * DDIM + (laneHi << 1);
            v2f A[32];
            #pragma unroll
            for (int kk = 0; kk < 32; ++kk)
                A[kk] = *(const v2f*)(xrow + kk * 4);

            float ep[8] = {0.f,0.f,0.f,0.f,0.f,0.f,0.f,0.f};

            #pragma unroll 1
            for (int j = 0; j < 8; ++j) {          // 8 column tiles of h
                const float evj = evec[j * 16 + m16];   // L2-hot scalar path
                const v2f* fp = (const v2f*)(sWf + j * 2048) + lane;
                v8f acc = {};
                #pragma unroll
                for (int kk = 0; kk < 32; ++kk) {  // chain K = 128 in steps of 4
                    const v2f b = fp[kk * 32];     // ds_load_b64, imm = kk*256
                    acc = __builtin_amdgcn_wmma_f32_16x16x4_f32(
                        false, A[kk], false, b, (short)0, acc, false, false);
                }
                // fuse: LeakyReLU(h) . e_vec  (h never leaves registers)
                #pragma unroll
                for (int v = 0; v < 8; ++v)
                    ep[v] = fmaf(lrelu(acc[v]), evj, ep[v]);
            }

            // sum over the 16 columns held by each 16-lane group
            #pragma unroll
            for (int v = 0; v < 8; ++v) {
                float s = ep[v];
                s += __shfl_xor(s, 1, 32);
                s += __shfl_xor(s, 2, 32);
                s += __shfl_xor(s, 4, 32);
                s += __shfl_xor(s, 8, 32);
                ep[v] = s;
            }
            // lane l wants row r = l&15 of this tile; rows 0-7 live in lanes<16
            // (ep[v] = row v), rows 8-15 in lanes>=16 (ep[v] = row v+8)
            float sel = ep[0];
            #pragma unroll
            for (int v = 1; v < 8; ++v)
                sel = ((m16 & 7) == v) ? ep[v] : sel;   // sel = ep[r & 7]
            const float other = __shfl_xor(sel, 16, 32);
            const float mine  = (laneHi == ((m16 >> 3) & 1)) ? sel : other;
            if (laneHi == half) edot = mine;   // half0 -> lanes 0-15, half1 -> 16-31
        }

        // ---- logits + softmax over the 32 neighbors (one per lane) ----
        float e = lrelu(edot + bias_l);
        float mx = e;
        #pragma unroll
        for (int d = 1; d < 32; d <<= 1) mx = fmaxf(mx, __shfl_xor(mx, d, 32));
        const float p = __expf(e - mx);
        float s = p;
        #pragma unroll
        for (int d = 1; d < 32; d <<= 1) s += __shfl_xor(s, d, 32);
        const float alpha = p * __builtin_amdgcn_rcpf(s);

        // ---- out[n,:] = sum_k alpha_k * labels[n,k,:]  (L = 64, 2 cols/lane) --
        float o0 = 0.f, o1 = 0.f;
        #pragma unroll
        for (int k = 0; k < 32; ++k) {
            const float ak = __uint_as_float(
                __builtin_amdgcn_readlane(__float_as_uint(alpha), k));
            o0 = fmaf(ak, lab[k * LDIM + lane], o0);
            o1 = fmaf(ak, lab[k * LDIM + lane + 32], o1);
        }
        out[(size_t)n * LDIM + lane]      = o0;
        out[(size_t)n * LDIM + lane + 32] = o1;
    }
}

extern "C" void kernel_launch(void* const* d_in, const int* in_sizes, int n_in,
                              void* d_out, int out_size, void* d_ws, size_t ws_size,
                              hipStream_t stream) {
    const float* X      = (const float*)d_in[0];  // para_neighbors [N,32,128]
    const float* labels = (const float*)d_in[1];  // para_nei_labels [N,32,64]
    const float* W      = (const float*)d_in[2];  // linear [128,128]
    const float* evec   = (const float*)d_in[3];  // e_vec [128]
    const float* bias   = (const float*)d_in[4];  // bias [32]
    float* out          = (float*)d_out;          // [N,64]

    const int N = in_sizes[0] / (KNB * DDIM);

    // 1024 blocks x 8 waves = 8192 waves, grid-stride over n; W staged once
    // per block (hot in L2 anyway, 192 MB).
    gat_fused_kernel<<<1024, 256, 0, stream>>>(X, labels, W, evec, bias, out, N);
}